// GraphVAE_43757126812205
// MI455X (gfx1250) — compile-verified
//
#include <hip/hip_runtime.h>

typedef float v2f __attribute__((ext_vector_type(2)));
typedef float v8f __attribute__((ext_vector_type(8)));

#define N_NODES 100000
#define F_DIM 128
#define D_LAT 64
#define N_EDGES 1000000
#define H_DIM 128

#define BLK 256
#define WPB 8                               // waves per block (wave32)
#define TILES_NODE (N_NODES / 16)           // 6250 (exact)
#define GRID_A ((TILES_NODE + WPB - 1) / WPB)    // 782
#define BATCHES_PER_SIDE (N_EDGES / 16)     // 62500
#define N_BATCH (2 * BATCHES_PER_SIDE)      // 125000
#define WAVES_B ((N_BATCH + 1) / 2)         // 62500 wave-jobs (2 batches each)
#define GRID_B ((WAVES_B + WPB - 1) / WPB)  // 7813
#define PADW 144                            // padded pair-row stride (v2f units):
                                            // 288 dwords % 64 banks = 32 -> kh=0/1
                                            // halves hit disjoint LDS bank ranges

// fp32 tensor op: D = A(16x4) * B(4x16) + C(16x16)
#define WMMA4(a, b, c) \
  __builtin_amdgcn_wmma_f32_16x16x4_f32(false, (a), false, (b), (short)0, (c), false, false)

// ---------------------------------------------------------------------------
// Kernel A: encoder. One wave computes a 16-row tile of mu[16,64], lv[16,64]
// via WMMA (K=128), then z = mu + eps*exp(0.5*lv) and KL partials.
// Weights staged in pair-interleaved LDS: one ds_load_b64 per B fragment.
// ---------------------------------------------------------------------------
__global__ __launch_bounds__(BLK) void vgae_encoder(
    const float* __restrict__ x, const float* __restrict__ eps,
    const float* __restrict__ W_mu, const float* __restrict__ b_mu,
    const float* __restrict__ W_lv, const float* __restrict__ b_lv,
    float* __restrict__ z, float* __restrict__ partial_kl)
{
  // [0 .. 4095] = W_mu pairs, [4096 .. 8191] = W_lv pairs ; 64KB total
  __shared__ v2f sWp[2 * (F_DIM / 2) * D_LAT];
  float* sWf = (float*)sWp;
  const int tid = threadIdx.x;
  for (int i = tid; i < F_DIM * D_LAT; i += BLK) {
    const int k = i >> 6, c = i & 63;
    const int pos = ((k >> 1) * D_LAT + c) * 2 + (k & 1);
    sWf[pos] = W_mu[i];
    sWf[(F_DIM / 2) * D_LAT * 2 + pos] = W_lv[i];
  }
  __syncthreads();

  const int wave = tid >> 5;
  const int lane = tid & 31;
  const int m = lane & 15;     // A-row / B-col within tile
  const int kh = lane >> 4;    // K-half select
  const int tile = blockIdx.x * WPB + wave;

  float kl = 0.0f;
  if (tile < TILES_NODE) {
    const int row0 = tile * 16;
    v8f cmu[4] = {};
    v8f clv[4] = {};

    const float* xrow = x + (size_t)(row0 + m) * F_DIM;
    for (int kb = 0; kb < F_DIM; kb += 4) {
      const v2f a = *(const v2f*)(xrow + kb + 2 * kh);  // A[m][kb+2kh .. +1]
      const int q = (kb >> 1) + kh;                     // k-pair index
      const v2f* bmu = sWp + q * D_LAT + m;
      const v2f* blv = bmu + (F_DIM / 2) * D_LAT;
#pragma unroll
      for (int t = 0; t < 4; ++t) {
        cmu[t] = WMMA4(a, bmu[16 * t], cmu[t]);
        clv[t] = WMMA4(a, blv[16 * t], clv[t]);
      }
    }

    // epilogue: C element r -> row = row0 + r + 8*kh, col = 16*t + m
#pragma unroll
    for (int t = 0; t < 4; ++t) {
      const int col = 16 * t + m;
      const float bm = b_mu[col];
      const float bl = b_lv[col];
#pragma unroll
      for (int r = 0; r < 8; ++r) {
        const int row = row0 + r + 8 * kh;
        const float mu = cmu[t][r] + bm;
        const float lv = clv[t][r] + bl;
        const float e = eps[(size_t)row * D_LAT + col];
        z[(size_t)row * D_LAT + col] = mu + e * expf(0.5f * lv);
        kl += 0.5f * (mu * mu + expf(lv) - 1.0f - lv);
      }
    }
  }

#pragma unroll
  for (int off = 16; off >= 1; off >>= 1) kl += __shfl_xor(kl, off, 32);
  if (lane == 0) partial_kl[(size_t)blockIdx.x * WPB + wave] = kl;
}

// ---------------------------------------------------------------------------
// Kernel B: edge scorer. One wave scores TWO 16-edge batches (B-fragment
// reuse: each ds_load_b64 feeds two WMMAs). Per batch:
//   P = zu*zv [16,64];  H = relu(P @ W1 + b1) [16,128];  logit = H @ W2 + b2
// ---------------------------------------------------------------------------
__global__ __launch_bounds__(BLK) void vgae_edges(
    const float* __restrict__ z,
    const float* __restrict__ W1, const float* __restrict__ b1,
    const float* __restrict__ W2, const float* __restrict__ b2,
    const int* __restrict__ pos_idx, const int* __restrict__ neg_idx,
    float* __restrict__ partial_recon)
{
  __shared__ v2f sW1p[(D_LAT / 2) * PADW];  // 36KB, bank-conflict-free b64 reads
  __shared__ float sW2[H_DIM];
  __shared__ float sb1[H_DIM];
  float* sW1f = (float*)sW1p;
  const int tid = threadIdx.x;
  for (int i = tid; i < D_LAT * H_DIM; i += BLK) {
    const int k = i >> 7, c = i & 127;
    sW1f[((k >> 1) * PADW + c) * 2 + (k & 1)] = W1[i];
  }
  if (tid < H_DIM) { sW2[tid] = W2[tid]; sb1[tid] = b1[tid]; }
  __syncthreads();

  const int wave = tid >> 5;
  const int lane = tid & 31;
  const int m = lane & 15;
  const int kh = lane >> 4;
  const int w = blockIdx.x * WPB + wave;   // wave-job id

  float ls = 0.0f;
  if (w < WAVES_B) {
    const int b0 = 2 * w;                  // two consecutive batches
    // gather bases for this lane's edge rows of each sub-batch
    const float* zp[4];                    // {zu0, zv0, zu1, zv1}
    bool pos_side[2];
#pragma unroll
    for (int s = 0; s < 2; ++s) {
      const int b = b0 + s;
      const bool is_pos = b < BATCHES_PER_SIDE;
      pos_side[s] = is_pos;
      const int* __restrict__ idx = is_pos ? pos_idx : neg_idx;
      const int e = (is_pos ? b : b - BATCHES_PER_SIDE) * 16 + m;
      zp[2 * s + 0] = z + (size_t)idx[e] * D_LAT;
      zp[2 * s + 1] = z + (size_t)idx[N_EDGES + e] * D_LAT;
    }

    v8f c[2][8] = {};                      // H pre-activation (2 batches x 8 tiles)
    for (int kb = 0; kb < D_LAT; kb += 4) {
      const int ko = kb + 2 * kh;
      const v2f a0 = (*(const v2f*)(zp[0] + ko)) * (*(const v2f*)(zp[1] + ko));
      const v2f a1 = (*(const v2f*)(zp[2] + ko)) * (*(const v2f*)(zp[3] + ko));
      const v2f* bp = sW1p + ((kb >> 1) + kh) * PADW + m;
#pragma unroll
      for (int t = 0; t < 8; ++t) {
        const v2f b = bp[16 * t];          // one ds_load_b64, reused twice
        c[0][t] = WMMA4(a0, b, c[0][t]);
        c[1][t] = WMMA4(a1, b, c[1][t]);
      }
    }

    const float b2v = b2[0];
#pragma unroll
    for (int s = 0; s < 2; ++s) {
      float p[8];
#pragma unroll
      for (int r = 0; r < 8; ++r) p[r] = 0.0f;
#pragma unroll
      for (int t = 0; t < 8; ++t) {
        const int col = 16 * t + m;
        const float bb = sb1[col];
        const float w2 = sW2[col];
#pragma unroll
        for (int r = 0; r < 8; ++r)
          p[r] += fmaxf(c[s][t][r] + bb, 0.0f) * w2;
      }
#pragma unroll
      for (int r = 0; r < 8; ++r) {        // reduce over the 16 column lanes
        p[r] += __shfl_xor(p[r], 8, 32);
        p[r] += __shfl_xor(p[r], 4, 32);
        p[r] += __shfl_xor(p[r], 2, 32);
        p[r] += __shfl_xor(p[r], 1, 32);
      }
      if (m == 0) {                        // lane 0: rows 0..7, lane 16: rows 8..15
        const float sgn = pos_side[s] ? 1.0f : -1.0f;
#pragma unroll
        for (int r = 0; r < 8; ++r) {
          const float l = sgn * (p[r] + b2v);
          ls += fminf(l, 0.0f) - log1pf(expf(-fabsf(l)));  // stable log_sigmoid
        }
      }
    }
    ls += __shfl_xor(ls, 16, 32);          // combine half-wave contributions
  }
  if (lane == 0) partial_recon[(size_t)blockIdx.x * WPB + wave] = ls;
}

// ---------------------------------------------------------------------------
// Final deterministic reduction -> scalar loss
// ---------------------------------------------------------------------------
__global__ __launch_bounds__(BLK) void vgae_finalize(
    const float* __restrict__ partial_recon, int n_recon,
    const float* __restrict__ partial_kl, int n_kl,
    float* __restrict__ out)
{
  __shared__ double sr[BLK];
  __shared__ double sk[BLK];
  const int tid = threadIdx.x;
  double r = 0.0, k = 0.0;
  for (int i = tid; i < n_recon; i += BLK) r += (double)partial_recon[i];
  for (int i = tid; i < n_kl; i += BLK) k += (double)partial_kl[i];
  sr[tid] = r; sk[tid] = k;
  __syncthreads();
  for (int s = BLK / 2; s > 0; s >>= 1) {
    if (tid < s) { sr[tid] += sr[tid + s]; sk[tid] += sk[tid + s]; }
    __syncthreads();
  }
  if (tid == 0) {
    const double recon = sr[0] / (double)(2 * N_EDGES);
    const double kl = sk[0] / (double)N_NODES;
    out[0] = (float)(kl - recon);  // -(recon - kl)
  }
}

extern "C" void kernel_launch(void* const* d_in, const int* in_sizes, int n_in,
                              void* d_out, int out_size, void* d_ws, size_t ws_size,
                              hipStream_t stream) {
  const float* x    = (const float*)d_in[0];
  const float* eps  = (const float*)d_in[1];
  const float* W_mu = (const float*)d_in[2];
  const float* b_mu = (const float*)d_in[3];
  const float* W_lv = (const float*)d_in[4];
  const float* b_lv = (const float*)d_in[5];
  const float* W1   = (const float*)d_in[6];
  const float* b1   = (const float*)d_in[7];
  const float* W2   = (const float*)d_in[8];
  const float* b2   = (const float*)d_in[9];
  const int* pos    = (const int*)d_in[10];
  const int* neg    = (const int*)d_in[11];

  float* z             = (float*)d_ws;                    // 6.4M floats = 25.6MB
  float* partial_kl    = z + (size_t)N_NODES * D_LAT;     // GRID_A*WPB = 6256
  float* partial_recon = partial_kl + 8192;               // GRID_B*WPB = 62504

  vgae_encoder<<<GRID_A, BLK, 0, stream>>>(x, eps, W_mu, b_mu, W_lv, b_lv,
                                           z, partial_kl);
  vgae_edges<<<GRID_B, BLK, 0, stream>>>(z, W1, b1, W2, b2, pos, neg,
                                         partial_recon);
  vgae_finalize<<<1, BLK, 0, stream>>>(partial_recon, GRID_B * WPB,
                                       partial_kl, GRID_A * WPB, (float*)d_out);
}